// HeteroGNN_54030688584319
// MI455X (gfx1250) — compile-verified
//
#include <hip/hip_runtime.h>
#include <hip/hip_bf16.h>
#include <stdint.h>

#define H 128
#define NUM_LAYERS 4
static const int N_LIG = 25000, N_PRO = 50000;
static const int E_LIG = 100000, E_PRO = 400000, E_INT = 200000;

typedef float v2f __attribute__((ext_vector_type(2)));
typedef float v8f __attribute__((ext_vector_type(8)));

// ---- monotone float<->uint keys for atomic segment-max ------------------
__device__ __forceinline__ unsigned f2key(float f) {
  unsigned u = __float_as_uint(f);
  return (u & 0x80000000u) ? ~u : (u | 0x80000000u);
}
__device__ __forceinline__ float key2f(unsigned k) {
  unsigned u = (k & 0x80000000u) ? (k ^ 0x80000000u) : ~k;
  return __uint_as_float(u);
}

// ---- small utility kernels ---------------------------------------------
__global__ void fill_f32(float* __restrict__ p, float v, long n) {
  long i = (long)blockIdx.x * 256 + threadIdx.x;
  if (i < n) p[i] = v;
}
__global__ void fill_u32(unsigned* __restrict__ p, unsigned v, long n) {
  long i = (long)blockIdx.x * 256 + threadIdx.x;
  if (i < n) p[i] = v;
}

// degree + segment-sum of edge attrs over dst (real edges only)
__global__ __launch_bounds__(256) void deg_easum_k(
    const int* __restrict__ dst, int dstOff,
    const float* __restrict__ ea, int D, int E,
    float* __restrict__ deg, float* __restrict__ easum) {
  const int e = blockIdx.x * 8 + (threadIdx.x >> 5);
  const int lane = threadIdx.x & 31;
  if (e >= E) return;
  const int d = dst[e] + dstOff;
  if (lane == 0) atomicAdd(deg + d, 1.0f);
  if (lane < D) atomicAdd(easum + (size_t)d * D + lane, ea[(size_t)e * D + lane]);
}

__global__ void loopea_k(const float* __restrict__ easum, const float* __restrict__ deg,
                         float* __restrict__ loopea, int N, int D) {
  long i = (long)blockIdx.x * 256 + threadIdx.x;
  if (i >= (long)N * D) return;
  int n = (int)(i / D);
  loopea[i] = easum[i] / fmaxf(deg[n], 1.0f);
}

// ---- fp32 WMMA GEMM: C[N x 128] = A[N x K] * W[K x 128] + bias ----------
// 256-thread block (8 waves) computes a 32 x 128 output slab. Each wave owns
// a 32x16 tile (two 16x16 accumulators) so every B fragment (strided W loads)
// feeds two V_WMMA_F32_16X16X4_F32 ops. Full blocks take a guard-free path.
__global__ __launch_bounds__(256) void gemm_bias_wmma(
    const float* __restrict__ A, const float* __restrict__ W,
    const float* __restrict__ bias, float* __restrict__ C, int N, int K) {
  const int lane = threadIdx.x & 31;
  const int wave = threadIdx.x >> 5;       // 0..7 -> column tile
  const int row0 = blockIdx.x * 32;
  const int col0 = wave * 16;
  const int half = lane >> 4;              // A: K+0/1 vs K+2/3 ; C: rows r vs r+8
  const int l = lane & 15;

  v8f acc0 = {}, acc1 = {};
  const float bv = bias[col0 + l];

  if (row0 + 32 <= N) {
    // ---- full path: no masks, no per-row guards ----
    const float* __restrict__ A0 = A + (size_t)(row0 + l) * K;
    const float* __restrict__ A1 = A + (size_t)(row0 + 16 + l) * K;
    if (row0 + 64 <= N) __builtin_prefetch(A0 + (size_t)32 * K, 0, 0);
    for (int k = 0; k < K; k += 4) {
      const int ka = k + half * 2;
      const float2 av0 = *(const float2*)(A0 + ka);
      const float2 av1 = *(const float2*)(A1 + ka);
      v2f a0, a1, b;
      a0.x = av0.x; a0.y = av0.y;
      a1.x = av1.x; a1.y = av1.y;
      b.x = W[(size_t)(ka + 0) * H + col0 + l];
      b.y = W[(size_t)(ka + 1) * H + col0 + l];
      acc0 = __builtin_amdgcn_wmma_f32_16x16x4_f32(false, a0, false, b,
                                                   (short)0, acc0, false, false);
      acc1 = __builtin_amdgcn_wmma_f32_16x16x4_f32(false, a1, false, b,
                                                   (short)0, acc1, false, false);
    }
#pragma unroll
    for (int r = 0; r < 8; ++r) {
      const int row = row0 + r + half * 8;
      C[(size_t)row * H + col0 + l] = acc0[r] + bv;
      C[(size_t)(row + 16) * H + col0 + l] = acc1[r] + bv;
    }
  } else {
    // ---- tail path: clamp rows, zero-mask A, guarded stores ----
    int r0 = row0 + l, r1 = row0 + 16 + l;
    float m0 = 1.0f, m1 = 1.0f;
    if (r0 >= N) { r0 = N - 1; m0 = 0.0f; }
    if (r1 >= N) { r1 = N - 1; m1 = 0.0f; }
    const float* __restrict__ A0 = A + (size_t)r0 * K;
    const float* __restrict__ A1 = A + (size_t)r1 * K;
    for (int k = 0; k < K; k += 4) {
      const int ka = k + half * 2;
      const float2 av0 = *(const float2*)(A0 + ka);
      const float2 av1 = *(const float2*)(A1 + ka);
      v2f a0, a1, b;
      a0.x = av0.x * m0; a0.y = av0.y * m0;
      a1.x = av1.x * m1; a1.y = av1.y * m1;
      b.x = W[(size_t)(ka + 0) * H + col0 + l];
      b.y = W[(size_t)(ka + 1) * H + col0 + l];
      acc0 = __builtin_amdgcn_wmma_f32_16x16x4_f32(false, a0, false, b,
                                                   (short)0, acc0, false, false);
      acc1 = __builtin_amdgcn_wmma_f32_16x16x4_f32(false, a1, false, b,
                                                   (short)0, acc1, false, false);
    }
#pragma unroll
    for (int r = 0; r < 8; ++r) {
      const int row = row0 + r + half * 8;
      if (row < N)      C[(size_t)row * H + col0 + l] = acc0[r] + bv;
      if (row + 16 < N) C[(size_t)(row + 16) * H + col0 + l] = acc1[r] + bv;
    }
  }
}

// ---- edge pass 1: logit + atomic segment-max ---------------------------
// One wave per (edge or self-loop); ea@We computed on the fly (D<=32).
__global__ __launch_bounds__(256) void edge_logit_k(
    const float* __restrict__ xl, const float* __restrict__ xr,
    const int* __restrict__ src, const int* __restrict__ dst, int srcOff, int dstOff,
    const float* __restrict__ ea, const float* __restrict__ loopea,
    const float* __restrict__ We, const float* __restrict__ att,
    float* __restrict__ logit, unsigned* __restrict__ lmaxkey,
    int E, int N, int D) {
  const int e = blockIdx.x * 8 + (threadIdx.x >> 5);
  const int lane = threadIdx.x & 31;
  if (e >= E + N) return;
  int s, d; const float* eap;
  if (e < E) { s = src[e] + srcOff; d = dst[e] + dstOff; eap = ea + (size_t)e * D; }
  else       { s = d = e - E;                            eap = loopea + (size_t)(e - E) * D; }
  float acc = 0.0f;
#pragma unroll
  for (int cc = 0; cc < 4; ++cc) {
    const int c = lane + cc * 32;
    float v = xl[(size_t)s * H + c] + xr[(size_t)d * H + c];
    for (int j = 0; j < D; ++j) v += eap[j] * We[(size_t)j * H + c];
    v = (v > 0.0f) ? v : 0.2f * v;          // leaky_relu(0.2)
    acc += v * att[c];
  }
#pragma unroll
  for (int off = 16; off > 0; off >>= 1) acc += __shfl_xor(acc, off, 32);
  if (lane == 0) {
    logit[e] = acc;
    atomicMax(lmaxkey + d, f2key(acc));
  }
}

// ---- edge pass 2: p = exp(logit - max), denom accumulate ---------------
__global__ void edge_p_k(const int* __restrict__ dst, int dstOff,
                         const float* __restrict__ logit, const unsigned* __restrict__ lmaxkey,
                         float* __restrict__ p, float* __restrict__ denom, int E, int N) {
  const int e = blockIdx.x * 256 + threadIdx.x;
  if (e >= E + N) return;
  const int d = (e < E) ? (dst[e] + dstOff) : (e - E);
  const float pv = __expf(logit[e] - key2f(lmaxkey[d]));
  p[e] = pv;
  atomicAdd(denom + d, pv);
}

// out[n] = bias (aggregation accumulates on top)
__global__ void init_out_k(float* __restrict__ out, const float* __restrict__ bias, int N) {
  long i = (long)blockIdx.x * 256 + threadIdx.x;
  if (i >= (long)N * H) return;
  out[i] = bias[i & (H - 1)];
}

// ---- edge pass 3: out[dst] += (p/denom) * xl[src] ----------------------
__global__ __launch_bounds__(256) void edge_aggr_k(
    const float* __restrict__ xl,
    const int* __restrict__ src, const int* __restrict__ dst, int srcOff, int dstOff,
    const float* __restrict__ p, const float* __restrict__ denom,
    float* __restrict__ out, int E, int N) {
  const int e = blockIdx.x * 8 + (threadIdx.x >> 5);
  const int lane = threadIdx.x & 31;
  if (e >= E + N) return;
  int s, d;
  if (e < E) { s = src[e] + srcOff; d = dst[e] + dstOff; }
  else       { s = d = e - E; }
  const float alpha = p[e] / denom[d];
#pragma unroll
  for (int cc = 0; cc < 4; ++cc) {
    const int c = lane + cc * 32;
    atomicAdd(out + (size_t)d * H + c, alpha * xl[(size_t)s * H + c]);
  }
}

// ---- host-side orchestration -------------------------------------------
struct ConvBufs {
  float *xl, *xr, *deg, *easum, *loopea, *logit, *p, *denom;
  unsigned* lmaxkey;
};

static void run_conv(hipStream_t st, const float* x, float* out, int N, int E, int D,
                     const int* src, const int* dst, int srcOff, int dstOff,
                     const float* ea,
                     const float* Wl, const float* bl, const float* Wr, const float* br,
                     const float* We, const float* att, const float* bias,
                     const ConvBufs& B) {
  const long ND = (long)N * D;
  fill_f32<<<(unsigned)((N + 255) / 256), 256, 0, st>>>(B.deg, 0.0f, N);
  fill_f32<<<(unsigned)((ND + 255) / 256), 256, 0, st>>>(B.easum, 0.0f, ND);
  fill_u32<<<(unsigned)((N + 255) / 256), 256, 0, st>>>(B.lmaxkey, 0u, N);
  fill_f32<<<(unsigned)((N + 255) / 256), 256, 0, st>>>(B.denom, 0.0f, N);

  deg_easum_k<<<(unsigned)((E + 7) / 8), 256, 0, st>>>(dst, dstOff, ea, D, E, B.deg, B.easum);
  loopea_k<<<(unsigned)((ND + 255) / 256), 256, 0, st>>>(B.easum, B.deg, B.loopea, N, D);

  gemm_bias_wmma<<<(unsigned)((N + 31) / 32), 256, 0, st>>>(x, Wl, bl, B.xl, N, H);
  gemm_bias_wmma<<<(unsigned)((N + 31) / 32), 256, 0, st>>>(x, Wr, br, B.xr, N, H);

  const int ET = E + N;
  edge_logit_k<<<(unsigned)((ET + 7) / 8), 256, 0, st>>>(B.xl, B.xr, src, dst, srcOff, dstOff,
                                                         ea, B.loopea, We, att,
                                                         B.logit, B.lmaxkey, E, N, D);
  edge_p_k<<<(unsigned)((ET + 255) / 256), 256, 0, st>>>(dst, dstOff, B.logit, B.lmaxkey,
                                                         B.p, B.denom, E, N);
  init_out_k<<<(unsigned)(((long)N * H + 255) / 256), 256, 0, st>>>(out, bias, N);
  edge_aggr_k<<<(unsigned)((ET + 7) / 8), 256, 0, st>>>(B.xl, src, dst, srcOff, dstOff,
                                                        B.p, B.denom, out, E, N);
}

extern "C" void kernel_launch(void* const* d_in, const int* in_sizes, int n_in,
                              void* d_out, int out_size, void* d_ws, size_t ws_size,
                              hipStream_t stream) {
  (void)in_sizes; (void)n_in; (void)out_size; (void)ws_size;
  const float* x_lig  = (const float*)d_in[0];
  const float* x_pro  = (const float*)d_in[1];
  const float* lig_ea = (const float*)d_in[2];
  const float* pro_ea = (const float*)d_in[3];
  const float* int_ea = (const float*)d_in[4];
  const int* lig_src = (const int*)d_in[5];
  const int* lig_dst = (const int*)d_in[6];
  const int* pro_src = (const int*)d_in[7];
  const int* pro_dst = (const int*)d_in[8];
  const int* int_src = (const int*)d_in[9];
  const int* int_dst = (const int*)d_in[10];
  // per-prefix params: Wl, bl, Wr, br, We, att, bias  (lig, pro, int)
  const float* P[21];
  for (int i = 0; i < 21; ++i) P[i] = (const float*)d_in[11 + i];

  const int NT = N_LIG + N_PRO;
  char* ws = (char*)d_ws;
  size_t off = 0;
  auto alloc = [&](size_t bytes) {
    void* ptr = ws + off;
    off += (bytes + 255) & ~(size_t)255;
    return ptr;
  };
  float* xc = (float*)alloc((size_t)NT * H * 4);   // current node features (concat)
  float* xn = (float*)alloc((size_t)NT * H * 4);   // next (lig/pro conv outputs)
  ConvBufs B;
  B.xl      = (float*)alloc((size_t)NT * H * 4);
  B.xr      = (float*)alloc((size_t)NT * H * 4);
  B.deg     = (float*)alloc((size_t)NT * 4);
  B.easum   = (float*)alloc((size_t)NT * 32 * 4);
  B.loopea  = (float*)alloc((size_t)NT * 32 * 4);
  B.lmaxkey = (unsigned*)alloc((size_t)NT * 4);
  B.denom   = (float*)alloc((size_t)NT * 4);
  const int EMAX = E_PRO + N_PRO;                  // 450k edges+loops max
  B.logit   = (float*)alloc((size_t)EMAX * 4);
  B.p       = (float*)alloc((size_t)EMAX * 4);

  hipMemcpyAsync(xc, x_lig, (size_t)N_LIG * H * 4, hipMemcpyDeviceToDevice, stream);
  hipMemcpyAsync(xc + (size_t)N_LIG * H, x_pro, (size_t)N_PRO * H * 4,
                 hipMemcpyDeviceToDevice, stream);

  for (int i = 0; i < NUM_LAYERS; ++i) {
    const size_t oW = (size_t)i * H * H, ob = (size_t)i * H;
    // ligand conv: rows [0, N_LIG)
    run_conv(stream, xc, xn, N_LIG, E_LIG, 16, lig_src, lig_dst, 0, 0, lig_ea,
             P[0] + oW, P[1] + ob, P[2] + oW, P[3] + ob,
             P[4] + (size_t)i * 16 * H, P[5] + ob, P[6] + ob, B);
    // protein conv: rows [N_LIG, NT)
    run_conv(stream, xc + (size_t)N_LIG * H, xn + (size_t)N_LIG * H, N_PRO, E_PRO, 32,
             pro_src, pro_dst, 0, 0, pro_ea,
             P[7] + oW, P[8] + ob, P[9] + oW, P[10] + ob,
             P[11] + (size_t)i * 32 * H, P[12] + ob, P[13] + ob, B);
    // interaction conv on concat graph; dst shifted by N_LIG
    run_conv(stream, xn, xc, NT, E_INT, 16, int_src, int_dst, 0, N_LIG, int_ea,
             P[14] + oW, P[15] + ob, P[16] + oW, P[17] + ob,
             P[18] + (size_t)i * 16 * H, P[19] + ob, P[20] + ob, B);
  }

  // reference returns (x_pro, x_lig)
  float* outp = (float*)d_out;
  hipMemcpyAsync(outp, xc + (size_t)N_LIG * H, (size_t)N_PRO * H * 4,
                 hipMemcpyDeviceToDevice, stream);
  hipMemcpyAsync(outp + (size_t)N_PRO * H, xc, (size_t)N_LIG * H * 4,
                 hipMemcpyDeviceToDevice, stream);
}